// TorchParametricSolver_81767587381594
// MI455X (gfx1250) — compile-verified
//
#include <hip/hip_runtime.h>

// ---------------- problem constants (match reference) ----------------
#define MDIM 256
#define NDIM 256
#define KDIM 256
#define NOPS (MDIM * NDIM * (KDIM + 1))   // 16,842,752
static_assert(NOPS == 16842752, "NOPS mismatch");

// ---------------- radix sort config ----------------
#define SORT_BLOCK 256
#define SORT_ITEMS 8
#define SORT_TILE  (SORT_BLOCK * SORT_ITEMS)   // 2048
#define NBLK       (NOPS / SORT_TILE)          // 8224, exact
static_assert(NBLK * SORT_TILE == NOPS, "tile must divide NOPS");

typedef __attribute__((ext_vector_type(2))) float v2f;
typedef __attribute__((ext_vector_type(8))) float v8f;

#if defined(__AMDGCN__) && defined(__gfx1250__) && \
    __has_builtin(__builtin_amdgcn_wmma_f32_16x16x4_f32)
#define USE_WMMA_RED 1
#else
#define USE_WMMA_RED 0
#endif

// Monotone map: float ordering -> unsigned ordering
__device__ __forceinline__ unsigned key_of(float f) {
    unsigned u = __float_as_uint(f);
    return u ^ ((u >> 31) ? 0xFFFFFFFFu : 0x80000000u);
}

// Block-wide sum of 256 per-thread partials, atomically added to *out.
// Uses V_WMMA_F32_16X16X4_F32 with a ones B-matrix: D[m][n] = sum_k A[m][k] + C,
// so 4 chained WMMAs reduce all 256 floats to 16 replicated row-sums (exact f32 RNE).
__device__ __forceinline__ void penalty_reduce_store(float v, float* sred, float* out) {
    const int tid = threadIdx.x;
    sred[tid] = v;
    __syncthreads();
#if USE_WMMA_RED
    if (tid < 32) {                       // whole wave 0 active: EXEC all ones
        const int lane = tid;
        const int m  = lane & 15;
        const int kb = (lane < 16) ? 0 : 2;   // A 16x4 layout: lanes 0-15 K0/K1, 16-31 K2/K3
        v2f bones;  bones.x = 1.0f; bones.y = 1.0f;
        v8f c = {};
#pragma unroll
        for (int ofs = 0; ofs < 256; ofs += 64) {
            v2f a;
            a.x = sred[ofs + m * 4 + kb + 0];
            a.y = sred[ofs + m * 4 + kb + 1];
            c = __builtin_amdgcn_wmma_f32_16x16x4_f32(
                    /*neg_a=*/false, a, /*neg_b=*/false, bones,
                    /*c_mod=*/(short)0, c, /*reuse_a=*/false, /*reuse_b=*/false);
        }
        // column n=0 lives on lane 0 (M=0..7 in c[0..7]) and lane 16 (M=8..15)
        float s = c[0] + c[1] + c[2] + c[3] + c[4] + c[5] + c[6] + c[7];
        s += __shfl(s, lane ^ 16, 32);
        if (lane == 0) atomicAdd(out, s);
    }
#else
    for (int off = 128; off > 0; off >>= 1) {
        if (tid < off) sred[tid] += sred[tid + off];
        __syncthreads();
    }
    if (tid == 0) atomicAdd(out, sred[0]);
#endif
}

// ---------------- kernels ----------------

__global__ void init_out_kernel(float* out) {
    if (threadIdx.x < 2) out[threadIdx.x] = 0.0f;
}

__global__ void make_keys_kernel(const float* __restrict__ opp,
                                 unsigned* __restrict__ keys,
                                 unsigned* __restrict__ idx) {
    size_t stride = (size_t)gridDim.x * blockDim.x;
    for (size_t i = (size_t)blockIdx.x * blockDim.x + threadIdx.x; i < (size_t)NOPS; i += stride) {
        keys[i] = key_of(opp[i]);
        idx[i]  = (unsigned)i;
    }
}

__global__ void radix_hist_kernel(const unsigned* __restrict__ keysIn,
                                  unsigned* __restrict__ blockHist,
                                  int shift) {
    __shared__ unsigned h[256];
    const int tid = threadIdx.x;
    const int b   = blockIdx.x;
    h[tid] = 0u;
    __syncthreads();
    const size_t base = (size_t)b * SORT_TILE;
#pragma unroll
    for (int k = 0; k < SORT_ITEMS; ++k) {
        unsigned key = keysIn[base + (size_t)k * SORT_BLOCK + tid];
        atomicAdd(&h[(key >> shift) & 255u], 1u);
    }
    __syncthreads();
    // digit-major: blockHist[d * NBLK + b]
    blockHist[(size_t)tid * NBLK + b] = h[tid];
}

__global__ void radix_scan_totals_kernel(const unsigned* __restrict__ blockHist,
                                         unsigned* __restrict__ digitBase) {
    __shared__ unsigned s[256];
    const int d = threadIdx.x;
    unsigned tot = 0u;
    const unsigned* row = blockHist + (size_t)d * NBLK;
    for (int b = 0; b < NBLK; ++b) tot += row[b];
    s[d] = tot;
    __syncthreads();
    for (int off = 1; off < 256; off <<= 1) {
        unsigned t = (d >= off) ? s[d - off] : 0u;
        __syncthreads();
        s[d] += t;
        __syncthreads();
    }
    digitBase[d] = s[d] - tot;   // exclusive
}

__global__ void radix_scan_blocks_kernel(unsigned* __restrict__ blockHist,
                                         const unsigned* __restrict__ digitBase) {
    __shared__ unsigned s[256];
    const int tid = threadIdx.x;
    const int d   = blockIdx.x;
    unsigned running = digitBase[d];
    unsigned* row = blockHist + (size_t)d * NBLK;
    for (int b0 = 0; b0 < NBLK; b0 += 256) {
        int b = b0 + tid;
        unsigned v = (b < NBLK) ? row[b] : 0u;
        s[tid] = v;
        __syncthreads();
        for (int off = 1; off < 256; off <<= 1) {
            unsigned t = (tid >= off) ? s[tid - off] : 0u;
            __syncthreads();
            s[tid] += t;
            __syncthreads();
        }
        if (b < NBLK) row[b] = running + (s[tid] - v);   // global start of (digit d, block b)
        unsigned tot = s[255];
        __syncthreads();
        running += tot;
    }
}

// Stable scatter: striped arrangement (element = blockStart + k*256 + tid),
// processed in (k, wave, lane) order == global order; within-wave ranks via ballots.
__global__ void radix_scatter_kernel(const unsigned* __restrict__ keysIn,
                                     const unsigned* __restrict__ idxIn,
                                     unsigned* __restrict__ keysOut,
                                     unsigned* __restrict__ idxOut,
                                     const unsigned* __restrict__ blockOfs,
                                     int shift) {
    __shared__ unsigned cnt[256];
    const int tid = threadIdx.x;
    const int b   = blockIdx.x;
    cnt[tid] = blockOfs[(size_t)tid * NBLK + b];
    __syncthreads();
    const size_t base   = (size_t)b * SORT_TILE;
    const int lane      = tid & 31;
    const int wv        = tid >> 5;
    const unsigned ltm  = (1u << lane) - 1u;
    for (int k = 0; k < SORT_ITEMS; ++k) {
        size_t i = base + (size_t)k * SORT_BLOCK + tid;
        unsigned key   = keysIn[i];
        unsigned pay   = idxIn[i];
        unsigned digit = (key >> shift) & 255u;
        for (int w = 0; w < SORT_BLOCK / 32; ++w) {
            if (wv == w) {
                unsigned mask = 0xFFFFFFFFu;
#pragma unroll
                for (int bit = 0; bit < 8; ++bit) {
                    unsigned vote = (unsigned)__ballot((digit >> bit) & 1u);
                    mask &= ((digit >> bit) & 1u) ? vote : ~vote;
                }
                int leader = __ffs(mask) - 1;
                unsigned pos = 0u;
                if (lane == leader) pos = atomicAdd(&cnt[digit], (unsigned)__popc(mask));
                pos = (unsigned)__shfl((int)pos, leader, 32);
                pos += (unsigned)__popc(mask & ltm);
                keysOut[pos] = key;
                idxOut[pos]  = pay;
            }
            __syncthreads();
        }
    }
}

// intra_pen is permutation-invariant: stream all rows of op_seq.
__global__ void intra_kernel(const float* __restrict__ mem,
                             const int* __restrict__ seq,
                             float* __restrict__ out) {
    __shared__ float sred[256];
    const size_t stride = (size_t)gridDim.x * blockDim.x;
    float acc = 0.0f;
    for (size_t i = (size_t)blockIdx.x * blockDim.x + threadIdx.x; i < (size_t)NOPS; i += stride) {
        if (i + 4 * stride < (size_t)NOPS)
            __builtin_prefetch(seq + 3 * (i + 4 * stride), 0, 1);   // global_prefetch_b8
        int i0 = seq[3 * i + 0];
        int i1 = seq[3 * i + 1];
        int i2 = seq[3 * i + 2];
        float a0 = mem[i0], a1 = mem[i1], a2 = mem[i2];
        float d1 = a1 - a0;
        float d2 = a2 - a1;
        float n1 = fminf(d1, 0.0f);
        float n2 = fminf(d2, 0.0f);
        acc += fmaxf(d1, 0.0f) + n1 * n1 + fmaxf(d2, 0.0f) + n2 * n2;
    }
    penalty_reduce_store(acc, sred, out + 1);
}

// inter_pen over consecutive sorted rows.
__global__ void inter_kernel(const float* __restrict__ mem,
                             const int* __restrict__ seq,
                             const unsigned* __restrict__ order,
                             float* __restrict__ out) {
    __shared__ float sred[256];
    const size_t stride = (size_t)gridDim.x * blockDim.x;
    float acc = 0.0f;
    for (size_t i = (size_t)blockIdx.x * blockDim.x + threadIdx.x; i + 1 < (size_t)NOPS; i += stride) {
        if (i + 4 * stride < (size_t)NOPS)
            __builtin_prefetch(order + i + 4 * stride, 0, 1);
        unsigned p = order[i];
        unsigned n = order[i + 1];
        float a2 = mem[seq[3 * (size_t)p + 2]];
        float a0 = mem[seq[3 * (size_t)n + 0]];
        float d  = a0 - a2;
        float nn = fminf(d, 0.0f);
        acc += fmaxf(d, 0.0f) + nn * nn;
    }
    penalty_reduce_store(acc, sred, out + 0);
}

// ---------------- launcher ----------------
extern "C" void kernel_launch(void* const* d_in, const int* in_sizes, int n_in,
                              void* d_out, int out_size, void* d_ws, size_t ws_size,
                              hipStream_t stream) {
    (void)in_sizes; (void)n_in; (void)out_size; (void)ws_size;
    const float* mem = (const float*)d_in[0];        // mem_params  (262144 f32)
    const float* opp = (const float*)d_in[1];        // op_params   (NOPS f32)
    const int*   seq = (const int*)d_in[2];          // op_seq      (NOPS*3 i32)
    float* out = (float*)d_out;                      // [0]=inter_pen, [1]=intra_pen

    // carve workspace
    char* w = (char*)d_ws;
    auto carve = [&w](size_t bytes) -> void* {
        void* p = (void*)w;
        w += (bytes + 255) & ~(size_t)255;
        return p;
    };
    unsigned* keysA = (unsigned*)carve((size_t)NOPS * 4);
    unsigned* keysB = (unsigned*)carve((size_t)NOPS * 4);
    unsigned* idxA  = (unsigned*)carve((size_t)NOPS * 4);
    unsigned* idxB  = (unsigned*)carve((size_t)NOPS * 4);
    unsigned* hist  = (unsigned*)carve((size_t)NBLK * 256 * 4);
    unsigned* dbase = (unsigned*)carve(256 * 4);

    init_out_kernel<<<1, 32, 0, stream>>>(out);
    make_keys_kernel<<<4096, 256, 0, stream>>>(opp, keysA, idxA);

    unsigned *ki = keysA, *ii = idxA, *ko = keysB, *io = idxB;
    for (int p = 0; p < 4; ++p) {
        const int shift = p * 8;
        radix_hist_kernel<<<NBLK, 256, 0, stream>>>(ki, hist, shift);
        radix_scan_totals_kernel<<<1, 256, 0, stream>>>(hist, dbase);
        radix_scan_blocks_kernel<<<256, 256, 0, stream>>>(hist, dbase);
        radix_scatter_kernel<<<NBLK, 256, 0, stream>>>(ki, ii, ko, io, hist, shift);
        unsigned* t;
        t = ki; ki = ko; ko = t;
        t = ii; ii = io; io = t;
    }
    // after 4 passes the sorted (key, index) pairs are back in keysA/idxA == ki/ii

    intra_kernel<<<4096, 256, 0, stream>>>(mem, seq, out);
    inter_kernel<<<4096, 256, 0, stream>>>(mem, seq, ii, out);
}